// InductiveMDGCNLayer_773094113324
// MI455X (gfx1250) — compile-verified
//
#include <hip/hip_runtime.h>

typedef __attribute__((ext_vector_type(2))) float v2f;
typedef __attribute__((ext_vector_type(8))) float v8f;

#define NN 4096   // nodes
#define FF 256    // features
#define DD 128    // embedding dim
#define KH 3      // hops (K+1)
#define CH 256    // K-chunk staged in LDS for the big SpMM

// D = A(16x4) * B(4x16) + C, fp32 WMMA, wave32.
__device__ __forceinline__ v8f wmma4(v2f a, v2f b, v8f c) {
  return __builtin_amdgcn_wmma_f32_16x16x4_f32(
      /*neg_a=*/false, a, /*neg_b=*/false, b,
      /*c_mod=*/(short)0, c, /*reuse_a=*/false, /*reuse_b=*/false);
}

// ---------------------------------------------------------------------------
// Kernel 1: E = X @ W_emb (mat 0) and XK[k] = X @ kernels[k] (mats 1..3).
// grid = (NN/16, 4), block = 256 (8 waves); wave w computes cols [16w,16w+16).
// ---------------------------------------------------------------------------
__global__ __launch_bounds__(256) void proj_kernel(
    const float* __restrict__ X, const float* __restrict__ W_emb,
    const float* __restrict__ kernels, float* __restrict__ E,
    float* __restrict__ XK) {
  const int lane = threadIdx.x & 31;
  const int wave = threadIdx.x >> 5;
  const int m0 = blockIdx.x * 16;
  const int n0 = wave * 16;
  const int mat = blockIdx.y;

  const float* Wm = (mat == 0) ? W_emb : kernels + (size_t)(mat - 1) * FF * DD;
  float* Op = (mat == 0) ? E : XK + (size_t)(mat - 1) * NN * DD;

  const int m = lane & 15;            // A row / B col / C col index
  const int half = lane >> 4;
  const int kh = 2 * half;

  v8f acc = {};
  const float* arow = X + (size_t)(m0 + m) * FF;
#pragma unroll 4
  for (int k0 = 0; k0 < FF; k0 += 4) {
    v2f a = *(const v2f*)(arow + k0 + kh);
    v2f b;
    b.x = Wm[(size_t)(k0 + kh) * DD + n0 + m];
    b.y = Wm[(size_t)(k0 + kh + 1) * DD + n0 + m];
    acc = wmma4(a, b, acc);
  }
  const int mb = 8 * half;
#pragma unroll
  for (int r = 0; r < 8; ++r)
    Op[(size_t)(m0 + mb + r) * DD + n0 + m] = acc[r];
}

// ---------------------------------------------------------------------------
// Kernel 2: low_rank = alpha * E @ E^T   [NN x NN], K = DD.
// grid = (NN/16, NN/128), block = 256 (8 waves, 16 cols each).
// B[k][n] = E[n][k] -> contiguous v2f load from E rows.
// ---------------------------------------------------------------------------
__global__ __launch_bounds__(256) void lowrank_kernel(
    const float* __restrict__ E, const float* __restrict__ alpha_p,
    float* __restrict__ lr) {
  const int lane = threadIdx.x & 31;
  const int wave = threadIdx.x >> 5;
  const int m0 = blockIdx.x * 16;
  const int n0 = blockIdx.y * 128 + wave * 16;
  const float alpha = alpha_p[0];

  const int m = lane & 15;
  const int half = lane >> 4;
  const int kh = 2 * half;

  const float* arow = E + (size_t)(m0 + m) * DD;
  const float* brow = E + (size_t)(n0 + m) * DD;  // E^T fragment = row of E

  v8f acc = {};
#pragma unroll 4
  for (int k0 = 0; k0 < DD; k0 += 4) {
    v2f a = *(const v2f*)(arow + k0 + kh);
    v2f b = *(const v2f*)(brow + k0 + kh);
    acc = wmma4(a, b, acc);
  }
  const int mb = 8 * half;
#pragma unroll
  for (int r = 0; r < 8; ++r)
    lr[(size_t)(m0 + mb + r) * NN + n0 + m] = alpha * acc[r];
}

// ---------------------------------------------------------------------------
// Kernel 3: per-row sums of (adj[k] + lr) for all 3 hops in one pass;
// dinv[k][row] = rsqrt(sum). grid = NN, block = 256.
// ---------------------------------------------------------------------------
__global__ __launch_bounds__(256) void rowsum_kernel(
    const float* __restrict__ adj, const float* __restrict__ lr,
    float* __restrict__ dinv) {
  __shared__ float red[4][256];
  const int row = blockIdx.x;
  const int tid = threadIdx.x;

  float s_lr = 0.f, s0 = 0.f, s1 = 0.f, s2 = 0.f;
  const float* lrow = lr + (size_t)row * NN;
  const float* a0 = adj + (size_t)row * NN;
  const float* a1 = a0 + (size_t)NN * NN;
  const float* a2 = a1 + (size_t)NN * NN;
  for (int c = tid; c < NN; c += 256) {
    s_lr += lrow[c];
    s0 += a0[c];
    s1 += a1[c];
    s2 += a2[c];
  }
  red[0][tid] = s_lr; red[1][tid] = s0; red[2][tid] = s1; red[3][tid] = s2;
  __syncthreads();
  for (int off = 128; off > 0; off >>= 1) {
    if (tid < off) {
#pragma unroll
      for (int j = 0; j < 4; ++j) red[j][tid] += red[j][tid + off];
    }
    __syncthreads();
  }
  if (tid == 0) {
    dinv[0 * NN + row] = rsqrtf(red[1][0] + red[0][0]);
    dinv[1 * NN + row] = rsqrtf(red[2][0] + red[0][0]);
    dinv[2 * NN + row] = rsqrtf(red[3][0] + red[0][0]);
  }
}

// ---------------------------------------------------------------------------
// Kernel 4: Y[k][j][d] = dinv[k][j] * XK[k][j][d]  (column-side diag scale).
// ---------------------------------------------------------------------------
__global__ __launch_bounds__(256) void scaley_kernel(
    const float* __restrict__ XK, const float* __restrict__ dinv,
    float* __restrict__ Y) {
  const size_t idx = (size_t)blockIdx.x * 256 + threadIdx.x;
  const int k = (int)(idx / ((size_t)NN * DD));
  const int j = (int)((idx / DD) % NN);
  Y[idx] = dinv[(size_t)k * NN + j] * XK[idx];
}

// ---------------------------------------------------------------------------
// Kernel 5: out = relu( sum_k diag(d_k) * (adj[k]+lr) @ Y[k] ).
// grid = NN/16 (M strips), block = 256 (8 waves cover the 128 output cols).
//
// LDS-staged, double-buffered software pipeline:
//   - the block cooperatively stages the summed A-strip (adj[hop]+lr) for a
//     16 x 256 K-chunk into LDS (global_load_b128 -> v_add -> ds_store_b128),
//     so each A element crosses the memory pipe once per block (not 8x),
//   - waves feed WMMA A-fragments from LDS via ds_load_b64,
//   - chunk c+1's global loads are issued before computing chunk c,
//   - one s_barrier per chunk (64 WMMAs); all control flow block-uniform,
//     so EXEC is all-ones at every WMMA (ISA 7.12 requirement).
// LDS use: 2 * 16 * 256 * 4B = 32 KB of the WGP's 320 KB.
// ---------------------------------------------------------------------------
__global__ __launch_bounds__(256) void spmm_kernel(
    const float* __restrict__ adj, const float* __restrict__ lr,
    const float* __restrict__ Y, const float* __restrict__ dinv,
    float* __restrict__ out) {
  const int tid = threadIdx.x;
  const int lane = tid & 31;
  const int wave = tid >> 5;
  const int m0 = blockIdx.x * 16;
  const int n0 = wave * 16;

  const int m = lane & 15;
  const int half = lane >> 4;
  const int kh = 2 * half;
  const int mb = 8 * half;

  // staging coordinates: 16 threads per row, 16 floats (4 x float4) each
  const int sr = tid >> 4;          // row 0..15 within the M strip
  const int sc = (tid & 15) * 16;   // column base within the K chunk

  __shared__ __align__(16) float As[2][16][CH];

  const size_t NN2 = (size_t)NN * NN;
  const float* lbase = lr + (size_t)(m0 + sr) * NN + sc;

  float4 sum[4];
  // ---- prologue: stage chunk 0 (hop 0, kbase 0) ----
  {
    const float* ap = adj + (size_t)(m0 + sr) * NN + sc;
#pragma unroll
    for (int i = 0; i < 4; ++i) {
      float4 av = ((const float4*)ap)[i];
      float4 lv = ((const float4*)lbase)[i];
      sum[i].x = av.x + lv.x; sum[i].y = av.y + lv.y;
      sum[i].z = av.z + lv.z; sum[i].w = av.w + lv.w;
    }
#pragma unroll
    for (int i = 0; i < 4; ++i) ((float4*)&As[0][sr][sc])[i] = sum[i];
  }
  __syncthreads();

  v8f total = {};
  v8f acc = {};
  const int NCHUNK = NN / CH;            // 16
  const int NCH_TOT = KH * NCHUNK;       // 48

  for (int cc = 0; cc < NCH_TOT; ++cc) {
    const int hop = cc >> 4;
    const int ch = cc & 15;
    const int kbase = ch * CH;
    const int buf = cc & 1;
    const bool more = (cc + 1) < NCH_TOT;   // block-uniform

    // ---- issue global loads for chunk cc+1 (overlap with WMMA below) ----
    if (more) {
      const int nhop = (cc + 1) >> 4;
      const int nch = (cc + 1) & 15;
      const float* ap =
          adj + (size_t)nhop * NN2 + (size_t)(m0 + sr) * NN + nch * CH + sc;
      const float* lp = lbase + nch * CH;
      __builtin_prefetch(ap + CH, 0, 1);   // global_prefetch_b8, one chunk ahead
      __builtin_prefetch(lp + CH, 0, 1);
#pragma unroll
      for (int i = 0; i < 4; ++i) {
        float4 av = ((const float4*)ap)[i];
        float4 lv = ((const float4*)lp)[i];
        sum[i].x = av.x + lv.x; sum[i].y = av.y + lv.y;
        sum[i].z = av.z + lv.z; sum[i].w = av.w + lv.w;
      }
    }

    // ---- 64 WMMAs on the staged chunk ----
    const float* Bm = Y + (size_t)hop * NN * DD + n0 + m;
    const float* as = &As[buf][m][0];
#pragma unroll 4
    for (int k0 = 0; k0 < CH; k0 += 4) {
      v2f a = *(const v2f*)(as + k0 + kh);       // ds_load_b64
      v2f b;
      b.x = Bm[(size_t)(kbase + k0 + kh) * DD];
      b.y = Bm[(size_t)(kbase + k0 + kh + 1) * DD];
      acc = wmma4(a, b, acc);
    }

    // ---- hop boundary: apply row scale d_k[i], reset accumulator ----
    if (ch == 15) {                         // block-uniform
      const float* dv = dinv + (size_t)hop * NN + m0 + mb;
#pragma unroll
      for (int r = 0; r < 8; ++r) {
        total[r] += dv[r] * acc[r];
        acc[r] = 0.f;
      }
    }

    // ---- commit chunk cc+1 into the other buffer ----
    if (more) {
#pragma unroll
      for (int i = 0; i < 4; ++i) ((float4*)&As[buf ^ 1][sr][sc])[i] = sum[i];
    }
    __syncthreads();
  }

#pragma unroll
  for (int r = 0; r < 8; ++r)
    out[(size_t)(m0 + mb + r) * DD + n0 + m] = fmaxf(total[r], 0.f);
}

// ---------------------------------------------------------------------------
extern "C" void kernel_launch(void* const* d_in, const int* in_sizes, int n_in,
                              void* d_out, int out_size, void* d_ws, size_t ws_size,
                              hipStream_t stream) {
  (void)in_sizes; (void)n_in; (void)out_size; (void)ws_size;
  const float* X       = (const float*)d_in[0];
  const float* adj     = (const float*)d_in[1];
  const float* W_emb   = (const float*)d_in[2];
  const float* kernels = (const float*)d_in[3];
  const float* alpha   = (const float*)d_in[4];
  float* out = (float*)d_out;

  // Workspace layout (floats): E | XK | Y | dinv | lr   (~82 MB total)
  float* ws   = (float*)d_ws;
  float* E    = ws;                          // NN*DD
  float* XK   = E + (size_t)NN * DD;         // KH*NN*DD
  float* Y    = XK + (size_t)KH * NN * DD;   // KH*NN*DD
  float* dinv = Y + (size_t)KH * NN * DD;    // KH*NN
  float* lr   = dinv + (size_t)KH * NN;      // NN*NN

  proj_kernel<<<dim3(NN / 16, KH + 1), 256, 0, stream>>>(X, W_emb, kernels, E, XK);
  lowrank_kernel<<<dim3(NN / 16, NN / 128), 256, 0, stream>>>(E, alpha, lr);
  rowsum_kernel<<<NN, 256, 0, stream>>>(adj, lr, dinv);
  scaley_kernel<<<(KH * NN * DD) / 256, 256, 0, stream>>>(XK, dinv, Y);
  spmm_kernel<<<NN / 16, 256, 0, stream>>>(adj, lr, Y, dinv, out);
}